// LightGCN_64965675319854
// MI455X (gfx1250) — compile-verified
//
#include <hip/hip_runtime.h>
#include <hip/hip_bf16.h>
#include <math.h>

#define N_USERS 200000
#define N_ITEMS 100000
#define N_NODES (N_USERS + N_ITEMS)
#define NNZ     4000000
#define DIM     64
#define N_LAYERS 3
#define BATCH   4096

typedef __attribute__((ext_vector_type(2))) float v2f;
typedef __attribute__((ext_vector_type(8))) float v8f;

// ---------------------------------------------------------------------------
// Init: cur = concat(user_emb, item_emb); acc = cur; nxt = 0.   (float4 lanes)
// ---------------------------------------------------------------------------
__global__ void lgcn_init_kernel(const float4* __restrict__ ue,
                                 const float4* __restrict__ ie,
                                 float4* __restrict__ cur,
                                 float4* __restrict__ acc,
                                 float4* __restrict__ nxt) {
  size_t i = (size_t)blockIdx.x * blockDim.x + threadIdx.x;
  const size_t n = (size_t)N_NODES * (DIM / 4);
  if (i >= n) return;
  const size_t ub = (size_t)N_USERS * (DIM / 4);
  float4 v = (i < ub) ? ue[i] : ie[i - ub];
  cur[i] = v;
  acc[i] = v;
  nxt[i] = make_float4(0.f, 0.f, 0.f, 0.f);
}

// ---------------------------------------------------------------------------
// SpMM: nxt[row] += val * cur[col].  16 lanes per edge, float4 per lane.
// cur+nxt together ~154MB -> resident in MI455X's 192MB L2, so both the
// random gather and the f32 atomic scatter resolve at L2 bandwidth.
// ---------------------------------------------------------------------------
__global__ void lgcn_spmm_kernel(const float* __restrict__ vals,
                                 const int*  __restrict__ rows,
                                 const int*  __restrict__ cols,
                                 const float4* __restrict__ cur,
                                 float* __restrict__ nxt) {
  long long t = (long long)blockIdx.x * blockDim.x + threadIdx.x;
  if (t >= (long long)NNZ * 16) return;
  int e = (int)(t >> 4);
  int q = (int)(t & 15);
  float v = vals[e];
  int r = rows[e];
  int c = cols[e];
  float4 x = cur[(size_t)c * 16 + q];
  float* dst = nxt + (size_t)r * DIM + q * 4;
  atomicAdd(dst + 0, v * x.x);   // global_atomic_add_f32, no-return
  atomicAdd(dst + 1, v * x.y);
  atomicAdd(dst + 2, v * x.z);
  atomicAdd(dst + 3, v * x.w);
}

// ---------------------------------------------------------------------------
// Fold: acc += nxt; cur = nxt; nxt = 0.
// ---------------------------------------------------------------------------
__global__ void lgcn_fold_kernel(float4* __restrict__ cur,
                                 float4* __restrict__ acc,
                                 float4* __restrict__ nxt) {
  size_t i = (size_t)blockIdx.x * blockDim.x + threadIdx.x;
  const size_t n = (size_t)N_NODES * (DIM / 4);
  if (i >= n) return;
  float4 x = nxt[i];
  float4 a = acc[i];
  a.x += x.x; a.y += x.y; a.z += x.z; a.w += x.w;
  acc[i] = a;
  cur[i] = x;
  nxt[i] = make_float4(0.f, 0.f, 0.f, 0.f);
}

// ---------------------------------------------------------------------------
// Loss: per wave32, one 16-row batch tile. Row-wise dots via
// V_WMMA_F32_16X16X4_F32 accumulating over K=64; diagonal of D gives scores.
//   A (16x4 f32): lanes 0-15 -> K=0,1 ; lanes 16-31 -> K=2,3 (documented)
//   B (4x16 f32): mirrored (lane = N, half-wave selects K pair)
//   D diag: i in 0..7  -> lane i,     c[i]
//           i in 8..15 -> lane 16+i,  c[i-8]
// ---------------------------------------------------------------------------
__device__ __forceinline__ float dsel8(v8f c, int r) {
  float x = c[0];
  x = (r == 1) ? c[1] : x;
  x = (r == 2) ? c[2] : x;
  x = (r == 3) ? c[3] : x;
  x = (r == 4) ? c[4] : x;
  x = (r == 5) ? c[5] : x;
  x = (r == 6) ? c[6] : x;
  x = (r == 7) ? c[7] : x;
  return x;
}

__device__ __forceinline__ float softplus_stable(float x) {
  return fmaxf(x, 0.f) + log1pf(expf(-fabsf(x)));
}

__global__ void lgcn_loss_kernel(const float* __restrict__ acc,
                                 const float* __restrict__ ue,
                                 const float* __restrict__ ie,
                                 const int* __restrict__ users,
                                 const int* __restrict__ pos,
                                 const int* __restrict__ neg,
                                 float* __restrict__ accum) {
  int tid  = blockIdx.x * blockDim.x + threadIdx.x;
  int lane = tid & 31;
  int tile = tid >> 5;          // grid sized so every tile index is valid
  int g    = lane >> 4;         // half-wave -> which K pair
  int mn   = lane & 15;         // row (A) / column (B) index in tile
  int bi   = tile * 16 + mn;

  int un = users[bi];
  int pn = pos[bi];
  int nn = neg[bi];

  const float* ur = acc + (size_t)un * DIM;
  const float* pr = acc + (size_t)(N_USERS + pn) * DIM;
  const float* nr = acc + (size_t)(N_USERS + nn) * DIM;

  v8f cpos = {};
  v8f cneg = {};
#pragma unroll
  for (int t = 0; t < 16; ++t) {
    int k = t * 4 + g * 2;
    v2f a;  a[0]  = ur[k]; a[1]  = ur[k + 1];
    v2f bp; bp[0] = pr[k]; bp[1] = pr[k + 1];
    v2f bn; bn[0] = nr[k]; bn[1] = nr[k + 1];
    cpos = __builtin_amdgcn_wmma_f32_16x16x4_f32(false, a, false, bp,
                                                 (short)0, cpos, false, false);
    cneg = __builtin_amdgcn_wmma_f32_16x16x4_f32(false, a, false, bn,
                                                 (short)0, cneg, false, false);
  }

  // Diagonal extraction (light_out = acc/4 => score scale 1/16).
  bool valid = (lane < 8) | (lane >= 24);
  int  rr    = (lane < 8) ? lane : ((lane >= 24) ? (lane - 24) : 0);
  float ps = dsel8(cpos, rr) * (1.0f / 16.0f);
  float ns = dsel8(cneg, rr) * (1.0f / 16.0f);
  float bpr = valid ? softplus_stable(ns - ps) : 0.f;

  // Regularization from layer-0 tables; each batch row appears on two lanes
  // (g=0,1), each covering half the 64 dims.
  const float* u0 = ue + (size_t)un * DIM;
  const float* p0 = ie + (size_t)pn * DIM;
  const float* n0 = ie + (size_t)nn * DIM;
  float reg = 0.f;
#pragma unroll 8
  for (int j = 0; j < 32; ++j) {
    int k = g * 32 + j;
    float a0 = u0[k], b0 = p0[k], c0 = n0[k];
    reg += a0 * a0 + b0 * b0 + c0 * c0;
  }

  // wave32 butterfly reduction
#pragma unroll
  for (int o = 16; o > 0; o >>= 1) {
    bpr += __shfl_xor(bpr, o, 32);
    reg += __shfl_xor(reg, o, 32);
  }
  if (lane == 0) {
    atomicAdd(accum + 0, bpr);
    atomicAdd(accum + 1, reg);
  }
}

__global__ void lgcn_zero2_kernel(float* a) { a[0] = 0.f; a[1] = 0.f; }

__global__ void lgcn_fin_kernel(const float* __restrict__ accum,
                                float* __restrict__ out) {
  out[0] = accum[0] / (float)BATCH;
  out[1] = 0.5f * accum[1] / (float)BATCH;
}

// ---------------------------------------------------------------------------
extern "C" void kernel_launch(void* const* d_in, const int* in_sizes, int n_in,
                              void* d_out, int out_size, void* d_ws, size_t ws_size,
                              hipStream_t stream) {
  const float* ue = (const float*)d_in[0];
  const float* ie = (const float*)d_in[1];
  const float* ev = (const float*)d_in[2];
  const int*   er = (const int*)d_in[3];
  const int*   ec = (const int*)d_in[4];
  const int*   us = (const int*)d_in[5];
  const int*   pi = (const int*)d_in[6];
  const int*   ni = (const int*)d_in[7];
  float* out = (float*)d_out;

  const size_t NB = (size_t)N_NODES * DIM;  // floats per node buffer
  float* cur   = (float*)d_ws;
  float* nxt   = cur + NB;
  float* acc   = nxt + NB;
  float* accum = acc + NB;                  // 2 floats

  const int blk = 256;

  lgcn_zero2_kernel<<<1, 1, 0, stream>>>(accum);

  const size_t n4 = (size_t)N_NODES * (DIM / 4);
  const int nb4 = (int)((n4 + blk - 1) / blk);
  lgcn_init_kernel<<<nb4, blk, 0, stream>>>((const float4*)ue, (const float4*)ie,
                                            (float4*)cur, (float4*)acc, (float4*)nxt);

  const long long sp_threads = (long long)NNZ * 16;
  const int sp_blocks = (int)((sp_threads + blk - 1) / blk);
  for (int l = 0; l < N_LAYERS; ++l) {
    lgcn_spmm_kernel<<<sp_blocks, blk, 0, stream>>>(ev, er, ec, (const float4*)cur, nxt);
    lgcn_fold_kernel<<<nb4, blk, 0, stream>>>((float4*)cur, (float4*)acc, (float4*)nxt);
  }

  // (BATCH/16) waves * 32 lanes = 8192 threads -> 32 blocks of 256
  lgcn_loss_kernel<<<(BATCH / 16) * 32 / blk, blk, 0, stream>>>(acc, ue, ie, us, pi, ni, accum);

  lgcn_fin_kernel<<<1, 1, 0, stream>>>(accum, out);
}